// FairEquiGenerator_33895881900475
// MI455X (gfx1250) — compile-verified
//
#include <hip/hip_runtime.h>
#include <hip/hip_bf16.h>
#include <cstdint>

// ---------------------------------------------------------------------------
// FairEquiGenerator for MI455X (gfx1250, wave32, WMMA f32 16x16x4)
//
// B=128, N=128, T=32, H=128, EMB=64, EXT=32
// Sequential LSTM+attention recurrence (8 blocks x 16 batch rows, all WMMA),
// then a full-device 256MB broadcast kernel for the (B, T*H, N) output.
// ---------------------------------------------------------------------------

typedef __attribute__((ext_vector_type(2))) float v2f;
typedef __attribute__((ext_vector_type(8))) float v8f;

#define Bb_ 128
#define Nn_ 128
#define T_  32
#define H_  128

// workspace layout (float offsets)
static constexpr size_t OFF_X    = 0;                        // x mean: 128*76 (74 padded to 76)
static constexpr size_t OFF_A    = OFF_X  + 128*76;          // a: 128*32
static constexpr size_t OFF_WQ   = OFF_A  + 128*32;          // Wq tiled: NT=16,KK=64 -> 65536
static constexpr size_t OFF_WV   = OFF_WQ + 16*64*64;        // Wv tiled: NT=8, KK=32 -> 16384
static constexpr size_t OFF_WG   = OFF_WV + 8*32*64;         // [Wih;Whh] tiled: NT=32,KK=105 -> 215040
static constexpr size_t OFF_BG   = OFF_WG + 32*105*64;       // combined gate bias: 512
static constexpr size_t OFF_MLP  = OFF_BG + 512;             // 7 tiled MLP layers
static constexpr size_t MLP_TOT  = (size_t)8*19*64 + 8*32*64 + 8*32*64 + 4*32*64 + 4*16*64 + 4*16*64 + 2*16*64;
static constexpr size_t OFF_PROJ = OFF_MLP + MLP_TOT;        // proj history: 8 blk * 32 t * 16 b * 128
static constexpr size_t OFF_HS   = OFF_PROJ + (size_t)8*32*16*128;  // sigmoid(h): 128*32*128
// total ~1.42M floats (~5.7 MB) of d_ws

__device__ __forceinline__ float sig_(float x) { return 1.0f / (1.0f + __expf(-x)); }

__device__ __forceinline__ v8f zero8() {
  v8f z = {0.f, 0.f, 0.f, 0.f, 0.f, 0.f, 0.f, 0.f};
  return z;
}

// D = A(16x4,f32) * B(4x16,f32) + C(16x16,f32)  -- v_wmma_f32_16x16x4_f32
__device__ __forceinline__ v8f wmma4(v2f a, v2f b, v8f c) {
  return __builtin_amdgcn_wmma_f32_16x16x4_f32(false, a, false, b, (short)0, c, false, false);
}

// ---------------------------------------------------------------------------
// Weight repack: W (Nc x Kc, row major) -> tiled B layout for 16x16x4 WMMA.
// Element (nt, kk, lane) holds v2f { W[n][k], W[n][k+1] } with
// n = nt*16 + (lane&15), k = 4*kk + 2*(lane>>4); zero-padded past Kc.
// ---------------------------------------------------------------------------
__global__ void prep_weight(const float* __restrict__ W, float* __restrict__ dst,
                            int Nc, int Kc, int KK) {
  int NT = Nc >> 4;
  int total = NT * KK * 32;
  for (int i = blockIdx.x * blockDim.x + threadIdx.x; i < total;
       i += gridDim.x * blockDim.x) {
    int lane = i & 31;
    int kk = (i >> 5) % KK;
    int nt = (i >> 5) / KK;
    int n = nt * 16 + (lane & 15);
    int k = 4 * kk + 2 * (lane >> 4);
    float x0 = (k     < Kc) ? W[(size_t)n * Kc + k    ] : 0.f;
    float x1 = (k + 1 < Kc) ? W[(size_t)n * Kc + k + 1] : 0.f;
    dst[2 * (size_t)i    ] = x0;
    dst[2 * (size_t)i + 1] = x1;
  }
}

// Fused [W_ih (512x290) ; W_hh (512x128)] -> 512 x 420 (padded) tiled layout.
__global__ void prep_gates(const float* __restrict__ Wih, const float* __restrict__ Whh,
                           float* __restrict__ dst) {
  const int KK = 105, NT = 32;
  int total = NT * KK * 32;
  for (int i = blockIdx.x * blockDim.x + threadIdx.x; i < total;
       i += gridDim.x * blockDim.x) {
    int lane = i & 31;
    int kk = (i >> 5) % KK;
    int nt = (i >> 5) / KK;
    int n = nt * 16 + (lane & 15);
    int kb = 4 * kk + 2 * (lane >> 4);
    float vals[2];
    for (int j = 0; j < 2; ++j) {
      int k = kb + j;
      float v = 0.f;
      if (k < 290)       v = Wih[(size_t)n * 290 + k];
      else if (k < 418)  v = Whh[(size_t)n * 128 + (k - 290)];
      vals[j] = v;
    }
    dst[2 * (size_t)i    ] = vals[0];
    dst[2 * (size_t)i + 1] = vals[1];
  }
}

__global__ void bias_comb(const float* __restrict__ a, const float* __restrict__ b,
                          float* __restrict__ d) {
  int i = blockIdx.x * blockDim.x + threadIdx.x;
  if (i < 512) d[i] = a[i] + b[i];
}

// ---------------------------------------------------------------------------
// x[b, 0:74] = mean over N of concat(node_emb[b], attrs[b]); cols 74,75 = 0 pad
// ---------------------------------------------------------------------------
__global__ __launch_bounds__(128) void mean_kernel(const float* __restrict__ node_emb,
                                                   const float* __restrict__ attrs,
                                                   float* __restrict__ x) {
  int b = blockIdx.x;
  int d = threadIdx.x;
  if (d < 76) {
    float s = 0.f;
    if (d < 64) {
      const float* p = node_emb + (size_t)b * Nn_ * 64 + d;
      for (int n = 0; n < Nn_; ++n) s += p[(size_t)n * 64];
    } else if (d < 74) {
      const float* p = attrs + (size_t)b * Nn_ * 10 + (d - 64);
      for (int n = 0; n < Nn_; ++n) s += p[(size_t)n * 10];
    }
    x[(size_t)b * 76 + d] = s * (1.0f / 128.0f);
  }
}

// ---------------------------------------------------------------------------
// 7-layer MLP, one wave per 16 batch rows, all layers via WMMA f32 16x16x4.
// dims: 74->128->128->128->64->64->64->32, ReLU each, final sigmoid.
// ---------------------------------------------------------------------------
__global__ __launch_bounds__(32) void mlp_kernel(float* __restrict__ ws,
    const float* __restrict__ b0, const float* __restrict__ b1,
    const float* __restrict__ b2, const float* __restrict__ b3,
    const float* __restrict__ b4, const float* __restrict__ b5,
    const float* __restrict__ b6) {
  __shared__ float bufA[16 * 128];
  __shared__ float bufB[16 * 128];
  const int tid  = threadIdx.x;           // one wave
  const int mn   = tid & 15;
  const int half = tid >> 4;
  const int blk  = blockIdx.x;

  const float* x = ws + OFF_X + (size_t)blk * 16 * 76;
  for (int i = tid; i < 16 * 76; i += 32) bufA[(i / 76) * 128 + (i % 76)] = x[i];
  __syncthreads();

  const int KKs[7]    = {19, 32, 32, 32, 16, 16, 16};
  const int NTs[7]    = {8, 8, 8, 4, 4, 4, 2};
  const size_t woff[7] = {0, 9728, 26112, 42496, 50688, 54784, 58880};
  const float* biases[7] = {b0, b1, b2, b3, b4, b5, b6};

  float* cur = bufA;
  float* nxt = bufB;
  for (int j = 0; j < 7; ++j) {
    const v2f* Bt = (const v2f*)(ws + OFF_MLP + woff[j]);
    const int KK = KKs[j], NT = NTs[j];
    const float* Ab = cur + mn * 128 + 2 * half;
    for (int nt = 0; nt < NT; ++nt) {
      v8f acc = zero8();
      for (int kk = 0; kk < KK; ++kk) {
        v2f a = *(const v2f*)(Ab + 4 * kk);
        acc = wmma4(a, Bt[(nt * KK + kk) * 32 + tid], acc);
      }
      float bb = biases[j][nt * 16 + mn];
      for (int r = 0; r < 8; ++r) {
        float z = fmaxf(acc[r] + bb, 0.f);
        if (j == 6) z = sig_(z);
        nxt[(r + 8 * half) * 128 + nt * 16 + mn] = z;
      }
    }
    __syncthreads();
    float* tmp = cur; cur = nxt; nxt = tmp;
  }
  float* aout = ws + OFF_A + (size_t)blk * 16 * 32;
  for (int i = tid; i < 16 * 32; i += 32) aout[i] = cur[(i >> 5) * 128 + (i & 31)];
}

// ---------------------------------------------------------------------------
// Sequential LSTM + history attention. 8 blocks x 256 threads (8 waves),
// each block owns a 16-row batch tile for all 32 steps.
// ---------------------------------------------------------------------------
__global__ __launch_bounds__(256) void lstm_kernel(float* __restrict__ ws,
    const float* __restrict__ pred, const float* __restrict__ bq,
    const float* __restrict__ bv) {
  __shared__ float kv[16 * 256];     // [b][0:128)=h, [128:256)=c
  __shared__ float qbuf[16 * 256];   // queries (2 heads x 128)
  __shared__ float inp[16 * 420];    // [s_att(256) | a_t(34) | h(128) | pad(2)]
  __shared__ float att[16 * 32 * 2]; // sigmoid logits -> attn weights

  const int tid  = threadIdx.x;
  const int w    = tid >> 5;         // wave 0..7
  const int lane = tid & 31;
  const int mn   = lane & 15;
  const int half = lane >> 4;
  const int blk  = blockIdx.x;

  const v2f* __restrict__ Bq = (const v2f*)(ws + OFF_WQ);
  const v2f* __restrict__ Bv = (const v2f*)(ws + OFF_WV);
  const v2f* __restrict__ Bg = (const v2f*)(ws + OFF_WG);
  const float* __restrict__ bg   = ws + OFF_BG;
  const float* __restrict__ avec = ws + OFF_A;
  float* __restrict__ projw = ws + OFF_PROJ + (size_t)blk * (T_ * 16 * H_);
  float* __restrict__ hsb   = ws + OFF_HS   + (size_t)blk * 16 * (T_ * H_);

  for (int i = tid; i < 16 * 256; i += 256) kv[i] = 0.f;
  if (tid < 16) { inp[tid * 420 + 418] = 0.f; inp[tid * 420 + 419] = 0.f; }
  __syncthreads();

  for (int t = 0; t < T_; ++t) {
    if (t == 0) {
      // step 0: s_att forced to zero
      for (int i = tid; i < 16 * 256; i += 256) inp[(i >> 8) * 420 + (i & 255)] = 0.f;
      __syncthreads();
    } else {
      // ---- queries = [h,c] @ Wq^T + bq : M=16, N=256 (2 tiles/wave), K=256
      {
        const float* Ab = &kv[mn * 256 + 2 * half];
        for (int i2 = 0; i2 < 2; ++i2) {
          int nt = w * 2 + i2;
          v8f acc = zero8();
          for (int kk = 0; kk < 64; ++kk) {
            v2f a = *(const v2f*)(Ab + 4 * kk);
            acc = wmma4(a, Bq[(nt * 64 + kk) * 32 + lane], acc);
          }
          float bqv = bq[nt * 16 + mn];
          for (int r = 0; r < 8; ++r)
            qbuf[(r + 8 * half) * 256 + nt * 16 + mn] = acc[r] + bqv;
        }
      }
      __syncthreads();
      // ---- logits[b,l,head] = <proj[l,b,:], q[b,head,:]>, then sigmoid
      {
        int P = 16 * t * 2;
        for (int idx = tid; idx < P; idx += 256) {
          int l = idx >> 5; int rem = idx & 31; int bb = rem >> 1; int head = rem & 1;
          const float* pr = projw + ((size_t)l * 16 + bb) * 128;
          const float* qr = &qbuf[bb * 256 + head * 128];
          float s = 0.f;
#pragma unroll 8
          for (int v = 0; v < 128; ++v) s += pr[v] * qr[v];
          att[(bb * 32 + l) * 2 + head] = sig_(s);
        }
      }
      __syncthreads();
      // ---- softmax over valid history rows l < t
      if (tid < 32) {
        int bb = tid >> 1, head = tid & 1;
        float mx = -1e30f;
        for (int l = 0; l < t; ++l) mx = fmaxf(mx, att[(bb * 32 + l) * 2 + head]);
        float den = 0.f;
        for (int l = 0; l < t; ++l) den += __expf(att[(bb * 32 + l) * 2 + head] - mx);
        float inv = 1.f / den;
        for (int l = 0; l < t; ++l)
          att[(bb * 32 + l) * 2 + head] = __expf(att[(bb * 32 + l) * 2 + head] - mx) * inv;
      }
      __syncthreads();
      // ---- s_att[b, 2v+head] = sum_l attn * proj[l,b,v] -> inp[:,0:256]
      for (int idx = tid; idx < 4096; idx += 256) {
        int bb = idx >> 8; int rem = idx & 255; int v = rem >> 1; int head = rem & 1;
        float s = 0.f;
        for (int l = 0; l < t; ++l)
          s += att[(bb * 32 + l) * 2 + head] * projw[((size_t)l * 16 + bb) * 128 + v];
        inp[bb * 420 + 2 * v + head] = s;
      }
      __syncthreads();
    }

    // ---- assemble a_t = [t, pred[b], a[b,:]] and old h into inp
    for (int idx = tid; idx < 16 * 34; idx += 256) {
      int bb = idx / 34, c = idx % 34;
      int gb = blk * 16 + bb;
      float v = (c == 0) ? (float)t : (c == 1) ? pred[gb] : avec[gb * 32 + (c - 2)];
      inp[bb * 420 + 256 + c] = v;
    }
    for (int idx = tid; idx < 16 * 128; idx += 256) {
      int bb = idx >> 7, hh = idx & 127;
      inp[bb * 420 + 290 + hh] = kv[bb * 256 + hh];
    }
    __syncthreads();

    // ---- gates = inp(16x420) @ [Wih;Whh]^T (420x512); wave w owns columns
    //      16w..16w+15 of all four gates (N-tiles w, w+8, w+16, w+24)
    float hn[8];
    {
      const float* Ab = &inp[mn * 420 + 2 * half];
      v8f aI = zero8(), aF = zero8(), aG = zero8(), aO = zero8();
      for (int kk = 0; kk < 105; ++kk) {
        v2f a = *(const v2f*)(Ab + 4 * kk);
        aI = wmma4(a, Bg[(((size_t)(w     ) * 105) + kk) * 32 + lane], aI);
        aF = wmma4(a, Bg[(((size_t)(w +  8) * 105) + kk) * 32 + lane], aF);
        aG = wmma4(a, Bg[(((size_t)(w + 16) * 105) + kk) * 32 + lane], aG);
        aO = wmma4(a, Bg[(((size_t)(w + 24) * 105) + kk) * 32 + lane], aO);
      }
      int hcol = w * 16 + mn;
      float bi = bg[hcol], bf = bg[128 + hcol], bgv = bg[256 + hcol], bo = bg[384 + hcol];
      for (int r = 0; r < 8; ++r) {
        int m = r + 8 * half;
        float cold  = kv[m * 256 + 128 + hcol];
        float igate = sig_(aI[r] + bi);
        float fgate = sig_(aF[r] + bf);
        float ggate = tanhf(aG[r] + bgv);
        float ogate = sig_(aO[r] + bo);
        float cnew  = fgate * cold + igate * ggate;
        float hne   = ogate * tanhf(cnew);
        kv[m * 256 + 128 + hcol] = cnew;   // this wave owns these columns
        kv[m * 256 + hcol]       = hne;
        hn[r] = hne;
        hsb[(size_t)m * (T_ * H_) + t * H_ + hcol] = sig_(hne);
      }
    }
    __syncthreads();
    // stage h_new as A-matrix for the proj GEMM
    {
      int hcol = w * 16 + mn;
      for (int r = 0; r < 8; ++r)
        inp[(r + 8 * half) * 420 + 290 + hcol] = hn[r];
    }
    __syncthreads();
    // ---- proj row t = h_new @ Wv^T + bv : M=16, N=128 (tile w), K=128
    {
      const float* Ab = &inp[mn * 420 + 290 + 2 * half];
      v8f acc = zero8();
      for (int kk = 0; kk < 32; ++kk) {
        v2f a = *(const v2f*)(Ab + 4 * kk);
        acc = wmma4(a, Bv[(w * 32 + kk) * 32 + lane], acc);
      }
      float bvv = bv[w * 16 + mn];
      for (int r = 0; r < 8; ++r)
        projw[((size_t)t * 16 + (r + 8 * half)) * 128 + w * 16 + mn] = acc[r] + bvv;
    }
    __threadfence();   // make proj/kv globally visible before next step's reads
    __syncthreads();
  }
}

// ---------------------------------------------------------------------------
// out[b, t*H+hh, n] = sigmoid(h_t[b,hh]) broadcast over n  (256 MB stream)
// ---------------------------------------------------------------------------
__global__ __launch_bounds__(256) void bcast_kernel(const float* __restrict__ hs,
                                                    float4* __restrict__ out) {
  size_t i = (size_t)blockIdx.x * 256 + threadIdx.x;  // 16,777,216 float4s
  if (i < (size_t)Bb_ * T_ * H_ * (Nn_ / 4)) {
    float v = hs[i >> 5];                             // 32 float4s per (b,t,hh) row
    out[i] = make_float4(v, v, v, v);
  }
}

// ---------------------------------------------------------------------------
extern "C" void kernel_launch(void* const* d_in, const int* in_sizes, int n_in,
                              void* d_out, int out_size, void* d_ws, size_t ws_size,
                              hipStream_t stream) {
  (void)in_sizes; (void)n_in; (void)out_size; (void)ws_size;
  const float* node_emb = (const float*)d_in[0];
  const float* attrs    = (const float*)d_in[1];
  const float* pred     = (const float*)d_in[2];
  const float* Wm[7]; const float* bm[7];
  for (int j = 0; j < 7; ++j) {
    Wm[j] = (const float*)d_in[3 + 2 * j];
    bm[j] = (const float*)d_in[4 + 2 * j];
  }
  const float* W_ih = (const float*)d_in[17];
  const float* W_hh = (const float*)d_in[18];
  const float* b_ih = (const float*)d_in[19];
  const float* b_hh = (const float*)d_in[20];
  const float* Wq   = (const float*)d_in[21];
  const float* bq   = (const float*)d_in[22];
  const float* Wv   = (const float*)d_in[23];
  const float* bv   = (const float*)d_in[24];

  float* ws  = (float*)d_ws;
  float* out = (float*)d_out;

  // repack weights into WMMA-tiled layouts (all reads hit L2 afterwards)
  prep_weight<<<256, 256, 0, stream>>>(Wq, ws + OFF_WQ, 256, 256, 64);
  prep_weight<<<256, 256, 0, stream>>>(Wv, ws + OFF_WV, 128, 128, 32);
  prep_gates <<<256, 256, 0, stream>>>(W_ih, W_hh, ws + OFF_WG);
  bias_comb  <<<2,   256, 0, stream>>>(b_ih, b_hh, ws + OFF_BG);
  {
    const int Ks[7]  = {74, 128, 128, 128, 64, 64, 64};
    const int Ns[7]  = {128, 128, 128, 64, 64, 64, 32};
    const int KKs[7] = {19, 32, 32, 32, 16, 16, 16};
    size_t off = OFF_MLP;
    for (int j = 0; j < 7; ++j) {
      prep_weight<<<256, 256, 0, stream>>>(Wm[j], ws + off, Ns[j], Ks[j], KKs[j]);
      off += (size_t)(Ns[j] >> 4) * KKs[j] * 64;
    }
  }

  mean_kernel<<<Bb_, 128, 0, stream>>>(node_emb, attrs, ws + OFF_X);
  mlp_kernel <<<8, 32, 0, stream>>>(ws, bm[0], bm[1], bm[2], bm[3], bm[4], bm[5], bm[6]);
  lstm_kernel<<<8, 256, 0, stream>>>(ws, pred, bq, bv);
  bcast_kernel<<<65536, 256, 0, stream>>>(ws + OFF_HS, (float4*)out);
}